// LocalMelSpecDiscretizer_16252156248527
// MI455X (gfx1250) — compile-verified
//
#include <hip/hip_runtime.h>
#include <stdint.h>

// CDNA5 WMMA operand vectors (f32 16x16x4: A/B = 2 VGPRs, C/D = 8 VGPRs)
typedef float v2f __attribute__((ext_vector_type(2)));
typedef float v8f __attribute__((ext_vector_type(8)));

#define NB      16
#define NT      4096
#define NC      80
#define NK      32
#define ROWS    64          // (b,t) rows per workgroup tile (4 x 16-row WMMA tiles)
#define TPC     4           // 16-row tiles per channel per workgroup
#define PAD     84          // padded LDS row stride (floats); stride-84 columns are bank-conflict-free
#define THREADS 256
#define WAVES   (THREADS / 32)
#define CH_PER_WAVE (NC / WAVES)   // 10

// AS3 -> flat cast: low 32 bits of the flat LDS aperture address are the LDS byte offset
__device__ __forceinline__ uint32_t lds_off(const void* p) {
  return (uint32_t)(uintptr_t)p;
}

__global__ __launch_bounds__(THREADS)
void melspec_discretize_kernel(const float* __restrict__ mel,
                               const float* __restrict__ cen,
                               float* __restrict__ out)
{
  __shared__ float sMel[ROWS * PAD];   // staged melspec tile (padded)
  __shared__ float sOut[ROWS * PAD];   // result tile (padded)
  __shared__ float sCent[NC * NK];     // full centroid table

  const uint32_t tid  = threadIdx.x;
  const uint32_t lane = tid & 31u;
  const uint32_t wave = tid >> 5;
  const uint32_t l16  = lane & 15u;
  const uint32_t hh   = lane >> 4;                 // 0: lanes 0-15, 1: lanes 16-31
  const uint32_t nothh = hh ^ 1u;                  // cross-half tie-break bit
  const uint64_t r0   = (uint64_t)blockIdx.x * ROWS;

  // ---- Stage inputs into LDS via CDNA5 async global->LDS DMA (ASYNCcnt) ----
  {
    const uint64_t gbase = (uint64_t)mel + r0 * (uint64_t)NC * 4ull; // tile rows are contiguous
    #pragma unroll
    for (int j = 0; j < (ROWS * NC) / THREADS; ++j) {                // 20 issues/thread
      uint32_t e = tid + j * THREADS;
      uint32_t r = e / NC, c = e - r * NC;
      uint32_t l = lds_off(&sMel[r * PAD + c]);
      uint32_t v = e * 4u;
      asm volatile("global_load_async_to_lds_b32 %0, %1, %2"
                   :: "v"(l), "v"(v), "s"(gbase) : "memory");
    }
    const uint64_t cbase = (uint64_t)cen;
    #pragma unroll
    for (int j = 0; j < (NC * NK) / THREADS; ++j) {                  // 10 issues/thread
      uint32_t e = tid + j * THREADS;
      uint32_t l = lds_off(&sCent[e]);
      uint32_t v = e * 4u;
      asm volatile("global_load_async_to_lds_b32 %0, %1, %2"
                   :: "v"(l), "v"(v), "s"(cbase) : "memory");
    }
    asm volatile("s_wait_asynccnt 0" ::: "memory");
  }
  __syncthreads();

  // Accumulator bias: score' = c*x - c^2/2 + 64 > 0 always (|x|<~6, |c|<~5),
  // so uint compare on the bits == float compare. Free via WMMA C operand.
  const v8f cbias = {64.0f, 64.0f, 64.0f, 64.0f, 64.0f, 64.0f, 64.0f, 64.0f};

  // ---- Per-channel: A operands built once, reused across TPC row-tiles ----
  // score[k][n] = c_k*x_n - 0.5*c_k^2 (+64); argmax_k == argmin_k |x_n - c_k|
  #pragma unroll 1
  for (int it = 0; it < CH_PER_WAVE; ++it) {
    const int ch = wave * CH_PER_WAVE + it;

    // A (16x4): row k -> [c_k, -c_k^2/2, 0, 0].
    // ISA 7.12.2: lanes 0-15 hold K=0 (v0), K=1 (v1); lanes 16-31 hold K=2,3 (zeros).
    float c0 = sCent[ch * NK + l16];
    float c1 = sCent[ch * NK + 16 + l16];
    v2f a0, a1;
    a0.x = hh ? 0.0f : c0;
    a0.y = hh ? 0.0f : -0.5f * c0 * c0;
    a1.x = hh ? 0.0f : c1;
    a1.y = hh ? 0.0f : -0.5f * c1 * c1;

    #pragma unroll 1
    for (int t = 0; t < TPC; ++t) {
      const uint32_t row = (uint32_t)t * 16u + l16;

      // B (4x16): row0 = x_n, row1 = 1, rows 2,3 = 0.
      float x = sMel[row * PAD + ch];
      v2f b;
      b.x = hh ? 1.0f : x;
      b.y = 0.0f;

      v8f d0 = __builtin_amdgcn_wmma_f32_16x16x4_f32(
          false, a0, false, b, (short)0, cbias, false, false);  // slot s=i   : k = hh*8+i
      v8f d1 = __builtin_amdgcn_wmma_f32_16x16x4_f32(
          false, a1, false, b, (short)0, cbias, false, false);  // slot s=8+i : k = 16+hh*8+i

      // Packed keys: [score bits 31:5 | (15-s)<<1 | 0]; max prefers the
      // highest score, then the smallest in-lane slot s (== smallest k).
      uint32_t u[16];
      #pragma unroll
      for (int i = 0; i < 8; ++i) {
        u[i]     = (__float_as_uint(d0[i]) & 0xFFFFFFE0u) | (uint32_t)((15 - i) << 1);
        u[i + 8] = (__float_as_uint(d1[i]) & 0xFFFFFFE0u) | (uint32_t)((7 - i) << 1);
      }
      // Log-depth unsigned-max tree (v_max_u32 / v_max3_u32, no serial chain).
      #pragma unroll
      for (int st = 8; st > 0; st >>= 1)
        #pragma unroll
        for (int i = 0; i < st; ++i)
          u[i] = (u[i] >= u[i + st]) ? u[i] : u[i + st];
      uint32_t key = u[0] | nothh;   // bit0: prefer half hh=0 on score tie

      // Combine with the other half-wave (lane ^ 16): SWAPX16 swizzle.
      uint32_t ko = (uint32_t)__builtin_amdgcn_ds_swizzle((int)key, 0x401F);
      key = (key >= ko) ? key : ko;

      // Decode winner: slot s, winning half -> k = hw*8 + s + (s>=8 ? 8 : 0)
      uint32_t sl  = 15u - ((key >> 1) & 15u);
      uint32_t hw8 = (key & 1u) ? 0u : 8u;
      uint32_t bk  = hw8 + sl + (sl & 8u);

      float val = sCent[ch * NK + bk];
      if (hh == 0) sOut[row * PAD + ch] = val;     // one writer per element
    }
  }
  __syncthreads();

  // ---- Coalesced write-back via async LDS -> global DMA ----
  {
    const uint64_t obase = (uint64_t)out + r0 * (uint64_t)NC * 4ull;
    #pragma unroll
    for (int j = 0; j < (ROWS * NC) / THREADS; ++j) {
      uint32_t e = tid + j * THREADS;
      uint32_t r = e / NC, c = e - r * NC;
      uint32_t l = lds_off(&sOut[r * PAD + c]);
      uint32_t v = e * 4u;
      asm volatile("global_store_async_from_lds_b32 %0, %1, %2"
                   :: "v"(v), "v"(l), "s"(obase) : "memory");
    }
    asm volatile("s_wait_asynccnt 0" ::: "memory");
  }
}

extern "C" void kernel_launch(void* const* d_in, const int* in_sizes, int n_in,
                              void* d_out, int out_size, void* d_ws, size_t ws_size,
                              hipStream_t stream) {
  const float* mel = (const float*)d_in[0];   // (B,T,C) f32
  const float* cen = (const float*)d_in[1];   // (C,K)  f32
  float* out = (float*)d_out;                 // (B,T,C) f32
  (void)in_sizes; (void)n_in; (void)out_size; (void)d_ws; (void)ws_size;

  const int nblocks = (NB * NT) / ROWS;       // 1024 workgroups, 64 rows each
  melspec_discretize_kernel<<<dim3(nblocks), dim3(THREADS), 0, stream>>>(mel, cen, out);
}